// SequentialParsingModel_20057497273118
// MI455X (gfx1250) — compile-verified
//
#include <hip/hip_runtime.h>

// Problem dims (fixed by the reference)
#define BB   16
#define SS   128
#define HH   512
#define LL   43
#define NEGV (-10000.0f)

typedef __attribute__((ext_vector_type(2))) float v2f;
typedef __attribute__((ext_vector_type(8))) float v8f;

// ---------------------------------------------------------------------------
// Kernel 1: projection GEMM  out[2048,512] = X[2048,512] @ W[0:512,:]
//           (+ optional tail: Xtail[2048,43] @ W[512:555,:]) + bias
// One wave32 computes one 16x16 f32 tile with V_WMMA_F32_16X16X4_F32.
// A-frag layout (16x4 f32): lane l -> row (l&15), K = k0 + 2*(l>>4) + {0,1}
// B-frag layout (4x16 f32): lane l -> col (l&15), K = k0 + 2*(l>>4) + {0,1}
// C/D layout  (16x16 f32): lane l, vgpr r -> M = r + 8*(l>>4), N = (l&15)
// ---------------------------------------------------------------------------
__global__ __launch_bounds__(256) void proj_gemm_wmma(
    const float* __restrict__ X,      // [2048, 512]
    const float* __restrict__ Xtail,  // [2048, 43] or nullptr
    const float* __restrict__ W,      // [Kfull, 512] row-major
    const float* __restrict__ bias,   // [512]
    float* __restrict__ out,          // [2048, 512]
    int useTail)
{
  const int lane  = threadIdx.x & 31;
  const int wave  = threadIdx.x >> 5;
  const int tile  = blockIdx.x * 8 + wave;  // 4096 tiles total
  const int tileM = tile >> 5;              // 2048/16 = 128 M-tiles
  const int tileN = tile & 31;              // 512/16  = 32  N-tiles
  const int rowBase = tileM * 16;
  const int colBase = tileN * 16;
  const int half = lane >> 4;               // 0 or 1
  const int mrow = lane & 15;

  v8f acc = {};
  const float* Arow = X + (rowBase + mrow) * HH;
  const float* Bcol = W + colBase + mrow;

#pragma unroll 4
  for (int k0 = 0; k0 < HH; k0 += 4) {
    const int ka = k0 + 2 * half;
    v2f a, b;
    a.x = Arow[ka];            // contiguous pair -> b64 load
    a.y = Arow[ka + 1];
    b.x = Bcol[(size_t)ka * HH];
    b.y = Bcol[(size_t)(ka + 1) * HH];
    // 8 args: (neg_a, A, neg_b, B, c_mod, C, reuse_a, reuse_b)
    acc = __builtin_amdgcn_wmma_f32_16x16x4_f32(
        false, a, false, b, (short)0, acc, false, false);
  }

  const int col = colBase + mrow;

  if (useTail) {
    // K = 512..554 : labels part of the concat, dense FMA tail.
    float s[8] = {0.f, 0.f, 0.f, 0.f, 0.f, 0.f, 0.f, 0.f};
    for (int l = 0; l < LL; ++l) {
      const float wv = W[(size_t)(HH + l) * HH + col];
#pragma unroll
      for (int r = 0; r < 8; ++r) {
        const int row = rowBase + half * 8 + r;
        s[r] += Xtail[(size_t)row * LL + l] * wv;
      }
    }
#pragma unroll
    for (int r = 0; r < 8; ++r) acc[r] += s[r];
  }

  const float bc = bias[col];
#pragma unroll
  for (int r = 0; r < 8; ++r) {
    const int row = rowBase + half * 8 + r;
    out[(size_t)row * HH + col] = acc[r] + bc;
  }
}

// ---------------------------------------------------------------------------
// Kernel 2: scores + masks + probs + log-softmax CE loss.
// One block per (b, i). w[b,i,:] and v staged in LDS; each wave owns a set
// of candidate heads j, does a 512-long dot with tanh via v_exp_f32/v_rcp_f32
// and a wave32 shuffle reduction.
// ---------------------------------------------------------------------------
__device__ __forceinline__ float fast_tanh(float x) {
  // tanh(x) = 1 - 2/(e^{2x}+1);  e^{2x} = exp2(2*log2(e)*x)
  const float e = __builtin_amdgcn_exp2f(x * 2.8853900817779268f);
  return 1.0f - 2.0f * __builtin_amdgcn_rcpf(e + 1.0f);
}

__device__ __forceinline__ float fast_exp(float x) {
  return __builtin_amdgcn_exp2f(x * 1.4426950408889634f);
}

__global__ __launch_bounds__(256) void score_softmax_kernel(
    const float* __restrict__ u,      // [B,S,H]
    const float* __restrict__ w,      // [B,S,H]
    const float* __restrict__ v,      // [H]
    const int*   __restrict__ heads,  // [B,S]
    const int*   __restrict__ words,  // [B,S]
    float* __restrict__ out)          // [0]=loss, [1..]=probs [B,S-1,S]
{
  __shared__ float sh_w[HH];
  __shared__ float sh_v[HH];
  __shared__ float sh_sc[SS];

  const int b    = blockIdx.x >> 7;   // / 128
  const int i    = blockIdx.x & 127;
  const int tid  = threadIdx.x;
  const int lane = tid & 31;
  const int wave = tid >> 5;

  for (int h = tid; h < HH; h += 256) {
    sh_w[h] = w[((size_t)b * SS + i) * HH + h];
    sh_v[h] = v[h];
  }
  __syncthreads();

  // scores over candidate heads j
  for (int j = wave; j < SS; j += 8) {
    const float* uj = u + ((size_t)b * SS + j) * HH;
    float p = 0.f;
#pragma unroll 4
    for (int t = 0; t < HH / 32; ++t) {
      const int h = lane + t * 32;
      p += sh_v[h] * fast_tanh(uj[h] + sh_w[h]);
    }
#pragma unroll
    for (int off = 16; off; off >>= 1) p += __shfl_xor(p, off, 32);
    if (lane == 0) {
      float sc = p;
      if (j == i || words[b * SS + j] == 0) sc = NEGV;  // diag + pad masks
      sh_sc[j] = sc;
    }
  }
  __syncthreads();

  // probs = exp(masked scores) for dependants i >= 1
  if (i >= 1) {
    const size_t obase = 1 + ((size_t)b * (SS - 1) + (i - 1)) * SS;
    for (int j = tid; j < SS; j += 256) out[obase + j] = fast_exp(sh_sc[j]);
  }

  // log-softmax over j + CE at heads[b,i]; accumulate loss = sum_i mean_b ce
  if (tid < 32) {
    float m = -3.0e38f;
    for (int j = lane; j < SS; j += 32) m = fmaxf(m, sh_sc[j]);
#pragma unroll
    for (int off = 16; off; off >>= 1) m = fmaxf(m, __shfl_xor(m, off, 32));
    float s = 0.f;
    for (int j = lane; j < SS; j += 32) s += fast_exp(sh_sc[j] - m);
#pragma unroll
    for (int off = 16; off; off >>= 1) s += __shfl_xor(s, off, 32);
    if (lane == 0 && i >= 1) {
      const int head = heads[b * SS + i];
      const float lse = m + __builtin_amdgcn_logf(s) * 0.6931471805599453f;
      const float ce  = lse - sh_sc[head];
      atomicAdd(out, ce * (1.0f / BB));
    }
  }
}

// ---------------------------------------------------------------------------
extern "C" void kernel_launch(void* const* d_in, const int* in_sizes, int n_in,
                              void* d_out, int out_size, void* d_ws, size_t ws_size,
                              hipStream_t stream) {
  (void)in_sizes; (void)n_in; (void)out_size; (void)ws_size;
  const float* repr   = (const float*)d_in[0];  // [B,S,H]
  const float* labels = (const float*)d_in[1];  // [B,S,L]
  const int*   heads  = (const int*)  d_in[2];  // [B,S]
  const int*   words  = (const int*)  d_in[3];  // [B,S]
  //            sent_id  = d_in[4] (unused)
  const float* u_w    = (const float*)d_in[5];  // [H+L,H]
  const float* u_b    = (const float*)d_in[6];  // [H]
  const float* w_w    = (const float*)d_in[7];  // [H+L,H]
  const float* w_b    = (const float*)d_in[8];  // [H]
  const float* v_w    = (const float*)d_in[9];  // [H,1]
  float* out = (float*)d_out;

  float* ws_u = (float*)d_ws;                      // [B*S, H] = 4 MB
  float* ws_w = ws_u + (size_t)BB * SS * HH;       // [B*S, H] = 4 MB

  // zero the loss accumulator (graph-capture safe)
  hipMemsetAsync(d_out, 0, sizeof(float), stream);

  // 4096 tiles, 8 waves/block -> 512 blocks
  proj_gemm_wmma<<<512, 256, 0, stream>>>(repr, nullptr, u_w, u_b, ws_u, 0);
  proj_gemm_wmma<<<512, 256, 0, stream>>>(repr, labels,  w_w, w_b, ws_w, 1);

  // one block per (b, i)
  score_softmax_kernel<<<BB * SS, 256, 0, stream>>>(ws_u, ws_w, v_w,
                                                    heads, words, out);
}